// EglGCN_20220706030037
// MI455X (gfx1250) — compile-verified
//
#include <hip/hip_runtime.h>

typedef __attribute__((ext_vector_type(2))) float v2f;
typedef __attribute__((ext_vector_type(8))) float v8f;

#define GCN_N 100000
#define GCN_D 64
#define WLDS_STRIDE 68   // 64 + 4 pad: lanes (stride 68 mod 64 = 4) land on distinct banks

// ---------------------------------------------------------------- zero fill
__global__ void gcn_zero4(float4* __restrict__ p, int n4) {
  int i = blockIdx.x * blockDim.x + threadIdx.x;
  if (i < n4) p[i] = make_float4(0.f, 0.f, 0.f, 0.f);
}

// ---------------------------------------------------------------- degrees
__global__ void gcn_degree(const int* __restrict__ esrc, const int* __restrict__ edst,
                           float* __restrict__ deg_out, float* __restrict__ deg_in, int E) {
  int e = blockIdx.x * blockDim.x + threadIdx.x;
  if (e >= E) return;
  atomicAdd(&deg_out[esrc[e]], 1.0f);
  atomicAdd(&deg_in[edst[e]], 1.0f);
}

// in-place deg -> rsqrt(max(deg,1))  (norm_src and norm_dst contiguous: 2N elems)
__global__ void gcn_norm(float* __restrict__ deg, int n) {
  int i = blockIdx.x * blockDim.x + threadIdx.x;
  if (i < n) deg[i] = rsqrtf(fmaxf(deg[i], 1.0f));
}

// ---------------------------------------------------------------- edge scatter
// one thread per (edge, 4-feature chunk): 16 threads per edge, d=64
__global__ void gcn_scatter(const float* __restrict__ h, const int* __restrict__ esrc,
                            const int* __restrict__ edst, const float* __restrict__ norm_src,
                            float* __restrict__ agg, int E) {
  unsigned tid = blockIdx.x * blockDim.x + threadIdx.x;
  unsigned e = tid >> 4;
  if (e >= (unsigned)E) return;
  int f = (tid & 15) << 2;
  int s = esrc[e];
  int d = edst[e];
  float ns = norm_src[s];
  float4 m = *(const float4*)(h + (size_t)s * GCN_D + f);
  float* ap = agg + (size_t)d * GCN_D + f;
  atomicAdd(ap + 0, m.x * ns);
  atomicAdd(ap + 1, m.y * ns);
  atomicAdd(ap + 2, m.z * ns);
  atomicAdd(ap + 3, m.w * ns);
}

// ---------------------------------------------------------------- WMMA GEMM
// out[N, LDW] = act( (agg * norm_dst[:,None]) @ W[64, LDW] + bias[:LDW] )
// One wave32 per 16x16 output tile, 16 chained V_WMMA_F32_16X16X4_F32 over K=64.
// W staged in LDS transposed + zero-padded to NCP cols -> branchless inner loop.
//   LDW: valid columns of W / out row stride (64 or 40)
//   NCP: padded columns, multiple of 16 (64 or 48)
//   NCT: column tiles = NCP/16
template <int LDW, int NCP, int NCT, bool RELU>
__global__ void gcn_gemm_wmma(const float* __restrict__ agg,
                              const float* __restrict__ norm_dst,
                              const float* __restrict__ W,
                              const float* __restrict__ bias,
                              float* __restrict__ out,
                              int nRowTiles) {
  __shared__ float Wlds[NCP * WLDS_STRIDE];

  // cooperative stage: Wlds[c*STRIDE + k] = W[k*LDW + c], zero for padded c
  for (int i = threadIdx.x; i < NCP * GCN_D; i += blockDim.x) {
    int c = i >> 6;          // / 64
    int k = i & 63;          // % 64
    Wlds[c * WLDS_STRIDE + k] = (c < LDW) ? W[k * LDW + c] : 0.f;
  }
  __syncthreads();

  int gw = blockIdx.x * (blockDim.x >> 5) + ((int)threadIdx.x >> 5);
  if (gw >= nRowTiles * NCT) return;   // uniform per-wave exit AFTER barrier

  int lane  = threadIdx.x & 31;
  int rt    = gw / NCT;
  int ct    = gw - rt * NCT;
  int r0    = rt << 4;

  // A-fragment (16x4 f32): lanes 0-15 hold K={0,1}+4k, lanes 16-31 K={2,3}+4k
  int mrow  = lane & 15;
  int khalf = (lane >> 4) << 1;        // 0 or 2
  int arow  = r0 + mrow;
  float nrm = norm_dst[arow];
  const float* aptr = agg + (size_t)arow * GCN_D + khalf;

  // B-fragment (4x16 f32): N = ct*16 + (lane&15), K striped like A
  int ncol = (ct << 4) + mrow;
  const float* bptr = &Wlds[ncol * WLDS_STRIDE + khalf];

  v8f acc = {0.f, 0.f, 0.f, 0.f, 0.f, 0.f, 0.f, 0.f};

#pragma unroll
  for (int ks = 0; ks < 16; ++ks) {
    v2f a, b;
    float2 av = *(const float2*)(aptr + (ks << 2));   // global_load_b64
    float2 bv = *(const float2*)(bptr + (ks << 2));   // ds_load_b64, bank-conflict-free
    a.x = av.x * nrm;
    a.y = av.y * nrm;
    b.x = bv.x;
    b.y = bv.y;
    acc = __builtin_amdgcn_wmma_f32_16x16x4_f32(
        /*neg_a=*/false, a, /*neg_b=*/false, b,
        /*c_mod=*/(short)0, acc, /*reuse_a=*/false, /*reuse_b=*/false);
  }

  bool colok = (LDW == NCP) || (ncol < LDW);
  float bb = colok ? bias[ncol] : 0.f;
  // C/D layout: VGPR i holds M = i + (lane>=16 ? 8 : 0), N = lane&15
  int mbase = r0 + ((lane >> 4) << 3);
#pragma unroll
  for (int i = 0; i < 8; ++i) {
    float v = acc[i] + bb;
    if (RELU) v = fmaxf(v, 0.f);
    if (colok) out[(size_t)(mbase + i) * LDW + ncol] = v;
  }
}

// ---------------------------------------------------------------- launch
extern "C" void kernel_launch(void* const* d_in, const int* in_sizes, int n_in,
                              void* d_out, int out_size, void* d_ws, size_t ws_size,
                              hipStream_t stream) {
  const float* features = (const float*)d_in[0];
  const int*   esrc     = (const int*)d_in[1];
  const int*   edst     = (const int*)d_in[2];
  const float* W1       = (const float*)d_in[3];
  const float* b1       = (const float*)d_in[4];
  const float* W2       = (const float*)d_in[5];
  const float* b2       = (const float*)d_in[6];
  const float* W3       = (const float*)d_in[7];
  const float* b3       = (const float*)d_in[8];
  float*       out      = (float*)d_out;

  const int N = GCN_N;
  const int E = in_sizes[1];

  // workspace layout (floats): [norm_src N][norm_dst N][agg N*64][h N*64]
  float* norm_src = (float*)d_ws;
  float* norm_dst = norm_src + N;
  float* agg      = norm_dst + N;
  float* h        = agg + (size_t)N * GCN_D;

  const int TPB = 256;

  // --- degrees -> norms ---
  {
    int n4 = (2 * N) / 4;
    gcn_zero4<<<(n4 + TPB - 1) / TPB, TPB, 0, stream>>>((float4*)norm_src, n4);
    gcn_degree<<<(E + TPB - 1) / TPB, TPB, 0, stream>>>(esrc, edst, norm_src, norm_dst, E);
    gcn_norm<<<(2 * N + TPB - 1) / TPB, TPB, 0, stream>>>(norm_src, 2 * N);
  }

  int aggN4      = (N * GCN_D) / 4;
  int zeroBlocks = (aggN4 + TPB - 1) / TPB;
  int scatBlocks = ((E * 16) + TPB - 1) / TPB;
  int rowTiles   = N / 16;                 // 6250, exact
  int wavesPB    = TPB / 32;               // 8

  // --- layer 1: features -> h ---
  gcn_zero4<<<zeroBlocks, TPB, 0, stream>>>((float4*)agg, aggN4);
  gcn_scatter<<<scatBlocks, TPB, 0, stream>>>(features, esrc, edst, norm_src, agg, E);
  {
    int total = rowTiles * 4;
    gcn_gemm_wmma<64, 64, 4, true><<<(total + wavesPB - 1) / wavesPB, TPB, 0, stream>>>(
        agg, norm_dst, W1, b1, h, rowTiles);
  }

  // --- layer 2: h -> h (gemm reads agg only; safe in-place on h) ---
  gcn_zero4<<<zeroBlocks, TPB, 0, stream>>>((float4*)agg, aggN4);
  gcn_scatter<<<scatBlocks, TPB, 0, stream>>>(h, esrc, edst, norm_src, agg, E);
  {
    int total = rowTiles * 4;
    gcn_gemm_wmma<64, 64, 4, true><<<(total + wavesPB - 1) / wavesPB, TPB, 0, stream>>>(
        agg, norm_dst, W2, b2, h, rowTiles);
  }

  // --- layer 3: h -> out [N, 40] (padded to 48 cols in LDS) ---
  gcn_zero4<<<zeroBlocks, TPB, 0, stream>>>((float4*)agg, aggN4);
  gcn_scatter<<<scatBlocks, TPB, 0, stream>>>(h, esrc, edst, norm_src, agg, E);
  {
    int total = rowTiles * 3;
    gcn_gemm_wmma<40, 48, 3, false><<<(total + wavesPB - 1) / wavesPB, TPB, 0, stream>>>(
        agg, norm_dst, W3, b3, out, rowTiles);
  }
}